// CTCLayer_25443386261809
// MI455X (gfx1250) — compile-verified
//
#include <hip/hip_runtime.h>
#include <cstdint>

// Problem constants (match reference)
#define Bn 1024
#define Tn 256
#define Cn 128
#define Ln 64
#define Sn (2 * Ln + 1)   // 129 extended states
#define BLANK (Cn - 1)    // 127
#define NEGF (-1e30f)
#define EPSF (1e-7f)

#define NTHREADS 160      // 5 wave32s: waves 0-3 issue row DMAs, threads 0..128 own states
#define NBUF 4            // async ring depth (prefetch distance 3)

// CDNA5 async copy: one DWORD per lane, global -> LDS, tracked by ASYNCcnt.
// VDST = per-lane LDS byte address, VADDR = per-lane 64-bit global address.
__device__ __forceinline__ void async_copy_f32(unsigned lds_byte_off, const float* g) {
    asm volatile("global_load_async_to_lds_b32 %0, %1, off"
                 :: "v"(lds_byte_off), "v"((unsigned long long)(uintptr_t)g)
                 : "memory");
}

// Per-wave async ops retire in order: waiting to <= N guarantees the oldest
// (queue depth - N) ops have written LDS.
__device__ __forceinline__ void wait_async_le(int n) {  // n is uniform
    if (n >= 2)      asm volatile("s_wait_asynccnt 0x2" ::: "memory");
    else if (n == 1) asm volatile("s_wait_asynccnt 0x1" ::: "memory");
    else             asm volatile("s_wait_asynccnt 0x0" ::: "memory");
}

__global__ __launch_bounds__(NTHREADS)
void ctc_alpha_kernel(const int* __restrict__ y_true,
                      const float* __restrict__ y_pred,
                      float* __restrict__ out) {
    __shared__ float prob[NBUF][Cn];     // ring of softmax rows (512 B each)
    __shared__ float alpha[2][Sn + 3];   // ping-pong alpha

    const int b   = blockIdx.x;
    const int tid = threadIdx.x;
    const float* base = y_pred + (size_t)b * Tn * Cn;

    // Per-state registers: class of ext[s] and skip-transition eligibility.
    int  cls     = BLANK;
    bool canskip = false;
    if (tid < Sn) {
        if (tid & 1) {                       // odd s -> label (s-1)/2
            const int j = tid >> 1;
            cls = y_true[b * Ln + j];
            if (tid >= 3) {
                const int prev = y_true[b * Ln + j - 1];
                canskip = (cls != prev);     // ext[s] != blank always for odd s
            }
        }
    }

    // Prologue: launch DMA for rows 0..2 (per wave: 3 outstanding async ops).
    if (tid < Cn) {
        async_copy_f32((unsigned)(uintptr_t)&prob[0][tid], base + 0 * Cn + tid);
        async_copy_f32((unsigned)(uintptr_t)&prob[1][tid], base + 1 * Cn + tid);
        async_copy_f32((unsigned)(uintptr_t)&prob[2][tid], base + 2 * Cn + tid);
    }

    // Time scan: one barrier per step; steady-state keeps 2 rows in flight.
    #pragma unroll 1
    for (int t = 0; t < Tn; ++t) {
        // Require row t landed; rows t+1, t+2 may still be in flight.
        wait_async_le(Tn - 1 - t);           // min(2, remaining) via clamp below
        __syncthreads();                     // row t + alpha(t-1) visible to all waves;
                                             // buf[(t-1)&3] no longer being read

        if ((t + 3) < Tn && tid < Cn)
            async_copy_f32((unsigned)(uintptr_t)&prob[(t + 3) & (NBUF - 1)][tid],
                           base + (size_t)(t + 3) * Cn + tid);

        if (tid < Sn) {
            const int   pb  = t & (NBUF - 1);
            const float lpv = __logf(prob[pb][cls] + EPSF);
            const int   cur = t & 1;
            const int   prv = cur ^ 1;
            float nv;
            if (t == 0) {
                // Only s=0 (blank) and s=1 (first label) reachable at t=0.
                nv = (tid < 2) ? lpv : NEGF;
            } else {
                const float a0 = alpha[prv][tid];
                const float a1 = (tid >= 1) ? alpha[prv][tid - 1] : NEGF;
                const float a2 = canskip    ? alpha[prv][tid - 2] : NEGF;
                const float m  = fmaxf(a0, fmaxf(a1, a2));
                const float s  = __expf(a0 - m) + __expf(a1 - m) + __expf(a2 - m);
                nv = m + __logf(s) + lpv;
            }
            alpha[cur][tid] = nv;
        }
    }

    __syncthreads();
    if (tid == 0) {
        const int fin = (Tn - 1) & 1;
        const float x = alpha[fin][Sn - 1];   // end on final blank
        const float y = alpha[fin][Sn - 2];   // or final label
        const float m = fmaxf(x, y);
        out[b] = -(m + __logf(__expf(x - m) + __expf(y - m)));
    }
}

extern "C" void kernel_launch(void* const* d_in, const int* in_sizes, int n_in,
                              void* d_out, int out_size, void* d_ws, size_t ws_size,
                              hipStream_t stream) {
    (void)in_sizes; (void)n_in; (void)d_ws; (void)ws_size; (void)out_size;
    const int*   y_true = (const int*)d_in[0];    // [B, L] labels
    const float* y_pred = (const float*)d_in[1];  // [B, T, C] softmax probs
    float*       out    = (float*)d_out;          // [B, 1] -> B floats

    ctc_alpha_kernel<<<Bn, NTHREADS, 0, stream>>>(y_true, y_pred, out);
}